// KBRD_48198122995881
// MI455X (gfx1250) — compile-verified
//
#include <hip/hip_runtime.h>
#include <math.h>

#define N_ENT 64368
#define DIMK  128
#define BATCH 512
#define SEQL  50

typedef float v2f __attribute__((ext_vector_type(2)));
typedef float v8f __attribute__((ext_vector_type(8)));

static __device__ __forceinline__ v8f wmma_f32(v2f a, v2f b, v8f c) {
    // D = A(16x4) x B(4x16) + C(16x16), f32, wave32
    return __builtin_amdgcn_wmma_f32_16x16x4_f32(
        /*neg_a=*/false, a, /*neg_b=*/false, b,
        /*c_mod=*/(short)0, c, /*reuse_a=*/false, /*reuse_b=*/false);
}

// ---------------------------------------------------------------------------
// Kernel A: w1q = q^T W1, w2q = q^T W2, qsum = sum(q); zero rowsum accumulators
// ---------------------------------------------------------------------------
__global__ __launch_bounds__(512) void KBRD_prep(
    const float* __restrict__ W1, const float* __restrict__ W2,
    const float* __restrict__ q,
    float* __restrict__ w1q, float* __restrict__ w2q,
    float* __restrict__ qsum, float* __restrict__ rowsum)
{
    const int t = threadIdx.x;
    if (t < DIMK) {
        float a1 = 0.f, a2 = 0.f;
        for (int e = 0; e < DIMK; ++e) {
            const float qe = q[e];
            a1 += qe * W1[e * DIMK + t];
            a2 += qe * W2[e * DIMK + t];
        }
        w1q[t] = a1;
        w2q[t] = a2;
    }
    if (t == 0) {
        float s = 0.f;
        for (int e = 0; e < DIMK; ++e) s += q[e];
        qsum[0] = s;
    }
    if (t < BATCH) rowsum[t] = 0.f;   // must re-zero every launch (graph replay)
}

// ---------------------------------------------------------------------------
// Kernel B: per-batch  v = gather + PE;  att = w1q.v + (w2q.vn + sb*qsum);
//           u_emb[b,:] = sum_l att[l] * v[l,:]
// ---------------------------------------------------------------------------
__global__ __launch_bounds__(128) void KBRD_uemb(
    const float* __restrict__ E, const int* __restrict__ seeds,
    const float* __restrict__ w1q, const float* __restrict__ w2q,
    const float* __restrict__ qsum, const float* __restrict__ soft_bias,
    float* __restrict__ u_emb)
{
    __shared__ float vbuf[SEQL * 129];   // padded stride vs bank conflicts
    __shared__ float w1s[DIMK];
    __shared__ float red[DIMK];
    __shared__ float att_s[SEQL];

    const int d = threadIdx.x;
    const int b = blockIdx.x;

    w1s[d] = w1q[d];
    // positional encoding frequency for this dim
    const float f = __expf(-(float)(d & ~1) * (logf(10000.0f) / (float)DIMK));
    for (int l = 0; l < SEQL; ++l) {
        const int idx = seeds[b * SEQL + l];
        const float ang = f * (float)l;
        const float pe  = ((d & 1) ? cosf(ang) : sinf(ang)) * 1e-3f;
        vbuf[l * 129 + d] = E[(size_t)idx * DIMK + d] + pe;
    }
    __syncthreads();

    // base = w2q . v[:,49] + soft_bias * qsum   (block reduction over d)
    red[d] = w2q[d] * vbuf[49 * 129 + d];
    __syncthreads();
    for (int s = 64; s > 0; s >>= 1) {
        if (d < s) red[d] += red[d + s];
        __syncthreads();
    }
    const float base = red[0] + soft_bias[0] * qsum[0];

    if (d < SEQL) {
        float s = 0.f;
        for (int k = 0; k < DIMK; ++k) s += w1s[k] * vbuf[d * 129 + k];
        att_s[d] = s + base;
    }
    __syncthreads();

    float acc = 0.f;
    for (int l = 0; l < SEQL; ++l) acc += att_s[l] * vbuf[l * 129 + d];
    u_emb[b * DIMK + d] = acc;
}

// ---------------------------------------------------------------------------
// Kernel C: scores = u_emb(512x128) @ E^T(128xN) + out_bias ; fused exp-sum
// Block: 256 threads (8 waves), one N-tile of 16 entities per block.
// Wave w: 4 M-tiles (rows w*64 .. w*64+63). K=128 -> 32 wmma steps per tile.
// ---------------------------------------------------------------------------
__global__ __launch_bounds__(256) void KBRD_scores(
    const float* __restrict__ u_emb, const float* __restrict__ E,
    const float* __restrict__ out_bias,
    float* __restrict__ scores, float* __restrict__ rowsum)
{
    __shared__ float Bt[16 * 132];   // 16 entity rows x 128 k, stride 132

    const int tid   = threadIdx.x;
    const int nbase = blockIdx.x * 16;

    // stage B tile: 16 rows x 128 floats = 512 float4, 2 per thread
    for (int it = 0; it < 2; ++it) {
        const int e  = it * 256 + tid;
        const int r  = e >> 5;
        const int c4 = (e & 31) << 2;
        const float4 v = *reinterpret_cast<const float4*>(
            &E[(size_t)(nbase + r) * DIMK + c4]);
        *reinterpret_cast<float4*>(&Bt[r * 132 + c4]) = v;   // 16B-aligned
    }
    __syncthreads();

    const int lane  = tid & 31;
    const int wave  = tid >> 5;
    const int n     = lane & 15;             // column within N-tile
    const int co    = (lane >> 4) << 1;      // k offset: 0 for lanes<16, 2 else
    const int mbase = wave << 6;             // 64 rows per wave

    v8f acc0 = {}, acc1 = {}, acc2 = {}, acc3 = {};

    const float* Bp = &Bt[n * 132 + co];
    const float* A0 = &u_emb[(size_t)(mbase +  0 + n) * DIMK + co];
    const float* A1 = &u_emb[(size_t)(mbase + 16 + n) * DIMK + co];
    const float* A2 = &u_emb[(size_t)(mbase + 32 + n) * DIMK + co];
    const float* A3 = &u_emb[(size_t)(mbase + 48 + n) * DIMK + co];

    #pragma unroll 4
    for (int k0 = 0; k0 < DIMK; k0 += 4) {
        const v2f bf = *reinterpret_cast<const v2f*>(Bp + k0);
        const v2f a0 = *reinterpret_cast<const v2f*>(A0 + k0);
        const v2f a1 = *reinterpret_cast<const v2f*>(A1 + k0);
        const v2f a2 = *reinterpret_cast<const v2f*>(A2 + k0);
        const v2f a3 = *reinterpret_cast<const v2f*>(A3 + k0);
        acc0 = wmma_f32(a0, bf, acc0);
        acc1 = wmma_f32(a1, bf, acc1);
        acc2 = wmma_f32(a2, bf, acc2);
        acc3 = wmma_f32(a3, bf, acc3);
    }

    // epilogue: bias add, store, per-row exp-sum (16-lane group reduce + atomic)
    const float bias = out_bias[nbase + n];
    const int   mo   = (lane >> 4) << 3;     // C layout: lanes>=16 hold M=j+8

    v8f accs[4] = {acc0, acc1, acc2, acc3};
    #pragma unroll
    for (int t = 0; t < 4; ++t) {
        #pragma unroll
        for (int j = 0; j < 8; ++j) {
            const int   m   = mbase + t * 16 + mo + j;
            const float val = accs[t][j] + bias;
            scores[(size_t)m * N_ENT + nbase + n] = val;
            float ex = __expf(val);
            ex += __shfl_xor(ex, 1, 16);
            ex += __shfl_xor(ex, 2, 16);
            ex += __shfl_xor(ex, 4, 16);
            ex += __shfl_xor(ex, 8, 16);
            if (n == 0) atomicAdd(&rowsum[m], ex);
        }
    }
}

// ---------------------------------------------------------------------------
// Kernel D: base_loss = mean_b( log(rowsum[b]) - scores[b, labels[b]] )
// (scores are tiny => unshifted logsumexp is numerically safe)
// ---------------------------------------------------------------------------
__global__ __launch_bounds__(512) void KBRD_loss(
    const float* __restrict__ rowsum, const int* __restrict__ labels,
    const float* __restrict__ scores, float* __restrict__ loss_out)
{
    __shared__ float red[BATCH];
    const int b = threadIdx.x;
    const float lz = logf(rowsum[b]);
    const float pk = scores[(size_t)b * N_ENT + labels[b]];
    red[b] = lz - pk;
    __syncthreads();
    for (int s = 256; s > 0; s >>= 1) {
        if (b < s) red[b] += red[b + s];
        __syncthreads();
    }
    if (b == 0) loss_out[0] = red[0] / (float)BATCH;
}

// ---------------------------------------------------------------------------
extern "C" void kernel_launch(void* const* d_in, const int* in_sizes, int n_in,
                              void* d_out, int out_size, void* d_ws, size_t ws_size,
                              hipStream_t stream)
{
    const float* entity_emb = (const float*)d_in[0];
    const float* W1         = (const float*)d_in[1];
    const float* W2         = (const float*)d_in[2];
    const float* q          = (const float*)d_in[3];
    const float* soft_bias  = (const float*)d_in[4];
    const float* out_bias   = (const float*)d_in[5];
    const int*   seed_sets  = (const int*)d_in[6];
    const int*   labels     = (const int*)d_in[7];

    float* out    = (float*)d_out;                 // [BATCH*N_ENT scores][1 loss]
    float* ws     = (float*)d_ws;
    float* w1q    = ws;                            // 128
    float* w2q    = ws + 128;                      // 128
    float* qsum   = ws + 256;                      // 1
    float* rowsum = ws + 272;                      // 512
    float* u_emb  = ws + 1024;                     // 512*128

    KBRD_prep<<<1, 512, 0, stream>>>(W1, W2, q, w1q, w2q, qsum, rowsum);
    KBRD_uemb<<<BATCH, 128, 0, stream>>>(entity_emb, seed_sets, w1q, w2q,
                                         qsum, soft_bias, u_emb);
    KBRD_scores<<<N_ENT / 16, 256, 0, stream>>>(u_emb, entity_emb, out_bias,
                                                out, rowsum);
    KBRD_loss<<<1, 512, 0, stream>>>(rowsum, labels, out,
                                     out + (size_t)BATCH * N_ENT);
}